// StructuredHingeLoss_1915555414697
// MI455X (gfx1250) — compile-verified
//
#include <hip/hip_runtime.h>
#include <stdint.h>

#define T_LEN    4096
#define MIN_GAP  5
#define MIN_SIZE 5
#define NTHREADS 256
#define CHUNK    (T_LEN / NTHREADS)   // 16 elements per thread

typedef unsigned int v4u __attribute__((ext_vector_type(4)));
typedef int          v8i __attribute__((ext_vector_type(8)));
typedef int          v4i __attribute__((ext_vector_type(4)));

// One workgroup per batch row. TDM DMAs the 16KB row into LDS, then a
// two-level suffix-max scan (smallest-j tie-break) + argmax reduction
// (smallest-i tie-break) reproduces jnp.argmax over the implicit TxT matrix.
__global__ __launch_bounds__(NTHREADS)
void best_span_kernel(const float* __restrict__ x_all, float* __restrict__ out, int n_batch)
{
    __shared__ float lds_x[T_LEN];
    __shared__ float lds_M[T_LEN];   // suffix max value
    __shared__ int   lds_J[T_LEN];   // suffix argmax (smallest j)
    __shared__ float red_v[NTHREADS];
    __shared__ int   red_i[NTHREADS];
    __shared__ int   red_j[NTHREADS];

    const int b = blockIdx.x;
    const int t = threadIdx.x;

    // ---------------- TDM: global row -> LDS (16 KB, contiguous) ----------------
    const uint64_t gaddr = (uint64_t)(uintptr_t)(x_all + (size_t)b * T_LEN);
    const uint32_t laddr = (uint32_t)(uintptr_t)(&lds_x[0]);

    // D# group 0: count=1 | lds_addr | global_addr[56:0] | type=2
    v4u g0;
    g0.x = 1u;
    g0.y = laddr;
    g0.z = (uint32_t)(gaddr & 0xffffffffull);
    g0.w = (uint32_t)((gaddr >> 32) & 0x1ffffffull) | (2u << 30);

    // D# group 1: data_size=4B, tensor_dim0=4096, tensor_dim1=1,
    //             tile_dim0=4096, tile_dim1=1, tile_dim2=0, dim0_stride=4096
    v8i g1;
    g1[0] = (int)(2u << 16);                               // data_size=2 (4 bytes)
    g1[1] = (int)(((uint32_t)T_LEN & 0xffffu) << 16);      // tensor_dim0 lo16
    g1[2] = (int)((((uint32_t)T_LEN) >> 16) | (1u << 16)); // dim0 hi16 | tensor_dim1=1
    g1[3] = (int)(((uint32_t)T_LEN & 0xffffu) << 16);      // dim1 hi16=0 | tile_dim0
    g1[4] = 1;                                             // tile_dim1=1, tile_dim2=0
    g1[5] = T_LEN;                                         // tensor_dim0_stride lo32
    g1[6] = 0;
    g1[7] = 0;

    // D# groups 2/3: unused dims set to 1, unused tiles to 0
    v4i g2; g2[0] = 1; g2[1] = 1; g2[2] = 0; g2[3] = 0;
    v4i g3; g3[0] = 0; g3[1] = (int)(1u << 16); g3[2] = 0; g3[3] = 0;

    if (t < 32) {                       // wave 0 issues the DMA and waits on it
#if defined(__clang_major__) && __clang_major__ >= 23
        v8i gz = (v8i)0;
        __builtin_amdgcn_tensor_load_to_lds(g0, g1, g2, g3, gz, 0);
#else
        __builtin_amdgcn_tensor_load_to_lds(g0, g1, g2, g3, 0);
#endif
        __builtin_amdgcn_s_wait_tensorcnt(0);
    }
    // The DMA wrote lds_x behind the compiler's back: escape the pointer with a
    // memory clobber so loads from lds_x are not folded away.
    { void* p = (void*)&lds_x[0]; asm volatile("" : "+v"(p) :: "memory"); }
    __syncthreads();

    // ---------------- per-chunk max with smallest-j tie-break ----------------
    const int base = t * CHUNK;
    {
        float cv = -__builtin_inff();
        int   cj = 0;
        #pragma unroll
        for (int k = CHUNK - 1; k >= 0; --k) {      // right->left: '>=' keeps smaller j
            float v = lds_x[base + k];
            if (v >= cv) { cv = v; cj = base + k; }
        }
        red_v[t] = cv; red_j[t] = cj;
    }
    __syncthreads();

    // ---------------- Hillis-Steele suffix max over the 256 chunk maxima ----------------
    for (int d = 1; d < NTHREADS; d <<= 1) {
        float ov = 0.0f; int oj = 0;
        const bool has = (t + d) < NTHREADS;
        if (has) { ov = red_v[t + d]; oj = red_j[t + d]; }
        __syncthreads();
        if (has && ov > red_v[t]) { red_v[t] = ov; red_j[t] = oj; }  // strict: left (smaller j) wins ties
        __syncthreads();
    }

    // ---------------- per-element suffix arrays M[k], J[k] ----------------
    {
        float rv; int rj;
        if (t + 1 < NTHREADS) { rv = red_v[t + 1]; rj = red_j[t + 1]; }
        else                  { rv = -__builtin_inff(); rj = 0; }
        #pragma unroll
        for (int k = CHUNK - 1; k >= 0; --k) {
            float v = lds_x[base + k];
            if (v >= rv) { rv = v; rj = base + k; }
            lds_M[base + k] = rv;
            lds_J[base + k] = rj;
        }
    }
    __syncthreads();

    // ---------------- candidates: v(i) = x[i] + M[i+MIN_SIZE], smallest-i tie-break ----------------
    {
        float bv = -__builtin_inff(); int bi = 0, bj = 0;
        #pragma unroll
        for (int k = 0; k < CHUNK; ++k) {           // ascending i: strict '>' keeps smaller i
            const int i = base + k;
            if (i >= MIN_GAP && (i + MIN_SIZE) < T_LEN) {
                float v = lds_x[i] + lds_M[i + MIN_SIZE];
                if (v > bv) { bv = v; bi = i; bj = lds_J[i + MIN_SIZE]; }
            }
        }
        red_v[t] = bv; red_i[t] = bi; red_j[t] = bj;
    }
    __syncthreads();

    // ---------------- tree reduction (lower thread index == smaller i wins ties) ----------------
    for (int s = NTHREADS >> 1; s > 0; s >>= 1) {
        if (t < s) {
            if (red_v[t + s] > red_v[t]) {
                red_v[t] = red_v[t + s];
                red_i[t] = red_i[t + s];
                red_j[t] = red_j[t + s];
            }
        }
        __syncthreads();
    }

    if (t == 0) {
        const float mval   = red_v[0];
        const float initv  = lds_x[1] + lds_x[1 + MIN_SIZE];
        const bool  better = mval > initv;
        out[b]               = better ? mval            : initv;
        out[n_batch + b]     = better ? (float)red_i[0] : 1.0f;
        out[2 * n_batch + b] = better ? (float)red_j[0] : (float)(1 + MIN_SIZE);
    }
}

extern "C" void kernel_launch(void* const* d_in, const int* in_sizes, int n_in,
                              void* d_out, int out_size, void* d_ws, size_t ws_size,
                              hipStream_t stream)
{
    (void)n_in; (void)out_size; (void)d_ws; (void)ws_size;
    const float* x   = (const float*)d_in[0];
    float*       out = (float*)d_out;
    const int n_batch = in_sizes[0] / T_LEN;   // B = 8
    best_span_kernel<<<dim3(n_batch), dim3(NTHREADS), 0, stream>>>(x, out, n_batch);
}